// YOLOv5_MobileNetV2_28673201668507
// MI455X (gfx1250) — compile-verified
//
#include <hip/hip_runtime.h>
#include <hip/hip_bf16.h>

typedef float v2f __attribute__((ext_vector_type(2)));
typedef float v8f __attribute__((ext_vector_type(8)));

#define NDET   1024
#define NBATCH 16
#define KTOP   100
#define CONF_T 0.4f
#define IOU_T  0.5f
#define NMS_T  256

// padded MLP dims
#define K1P 160   // 147 -> 160
#define N1P 256   // 255 -> 256
#define N2  128
#define N3P 16    // 4 -> 16
#define LDA 164   // LDS stride for crops stripe (16 x 160)
#define LDF 260   // LDS stride for feat (16 x 256)
#define LDH 132   // LDS stride for hidden (16 x 128)

// ---------------------------------------------------------------------------
// Kernel 0: pad/pack weights + biases into workspace
// ---------------------------------------------------------------------------
__global__ __launch_bounds__(256) void pack_kernel(
    const float* __restrict__ W1, const float* __restrict__ b1,
    const float* __restrict__ W2, const float* __restrict__ b2,
    const float* __restrict__ W3, const float* __restrict__ b3,
    float* __restrict__ W1p, float* __restrict__ W2p, float* __restrict__ W3p,
    float* __restrict__ b1p, float* __restrict__ b2p, float* __restrict__ b3p)
{
    int i = blockIdx.x * blockDim.x + threadIdx.x;
    const int e1 = K1P * N1P;            // 40960
    const int e2 = e1 + N1P * N2;        // +32768
    const int e3 = e2 + N2 * N3P;        // +2048
    const int e4 = e3 + N1P;             // +256
    const int e5 = e4 + N2;              // +128
    const int e6 = e5 + N3P;             // +16
    if (i < e1) {
        int k = i / N1P, n = i % N1P;
        W1p[i] = (k < 147 && n < 255) ? W1[k * 255 + n] : 0.0f;
    } else if (i < e2) {
        int r = i - e1; int k = r / N2, n = r % N2;
        W2p[r] = (k < 255) ? W2[k * N2 + n] : 0.0f;
    } else if (i < e3) {
        int r = i - e2; int k = r / N3P, n = r % N3P;
        W3p[r] = (n < 4) ? W3[k * 4 + n] : 0.0f;
    } else if (i < e4) {
        int n = i - e3;
        b1p[n] = (n < 255) ? b1[n] : 0.0f;
    } else if (i < e5) {
        b2p[i - e4] = b2[i - e4];
    } else if (i < e6) {
        int n = i - e5;
        b3p[n] = (n < 4) ? b3[n] : 0.0f;
    }
}

// ---------------------------------------------------------------------------
// Kernel 1: per-batch NMS (bitonic sort in LDS + sequential suppression scan)
// one block per batch; 256 threads
// ---------------------------------------------------------------------------
__global__ __launch_bounds__(NMS_T) void nms_kernel(
    const float* __restrict__ dets,      // [16][1024][6]
    float* __restrict__ selBoxes,        // [16][100][4]
    float* __restrict__ validW)          // [16][100]
{
    __shared__ float skey[NDET];
    __shared__ int   sidx[NDET];
    __shared__ float sx1[NDET], sy1[NDET], sx2[NDET], sy2[NDET];
    __shared__ int   supp[NDET];

    const int tid   = threadIdx.x;
    const int batch = blockIdx.x;
    const float* db = dets + (size_t)batch * NDET * 6;

    for (int i = tid; i < NDET; i += NMS_T) { skey[i] = db[i * 6 + 4]; sidx[i] = i; }

    // bitonic sort: descending score, ties by ascending original index (stable argsort(-s))
    for (int k = 2; k <= NDET; k <<= 1) {
        for (int j = k >> 1; j > 0; j >>= 1) {
            __syncthreads();
            for (int i = tid; i < NDET; i += NMS_T) {
                int ixj = i ^ j;
                if (ixj > i) {
                    float ka = skey[i], kb = skey[ixj];
                    int   ia = sidx[i], ib = sidx[ixj];
                    bool aAfterB = (ka < kb) || (ka == kb && ia > ib);
                    bool desc = ((i & k) == 0);
                    if (desc ? aAfterB : !aAfterB) {
                        skey[i] = kb; skey[ixj] = ka;
                        sidx[i] = ib; sidx[ixj] = ia;
                    }
                }
            }
        }
    }
    __syncthreads();

    // gather sorted boxes (xywh -> xyxy), init suppression flags
    for (int i = tid; i < NDET; i += NMS_T) {
        const float* d = db + sidx[i] * 6;
        float cx = d[0], cy = d[1], hw = d[2] * 0.5f, hh = d[3] * 0.5f;
        sx1[i] = cx - hw; sy1[i] = cy - hh; sx2[i] = cx + hw; sy2[i] = cy + hh;
        supp[i] = 0;
    }
    __syncthreads();

    // sequential over pivot i (uniform branches), parallel over j
    for (int i = 0; i < NDET; ++i) {
        float si = skey[i];
        if (si <= CONF_T) break;         // sorted: nothing below can suppress
        if (supp[i]) continue;           // uniform read; no writes this iter
        float ix1 = sx1[i], iy1 = sy1[i], ix2 = sx2[i], iy2 = sy2[i];
        float iarea = (ix2 - ix1) * (iy2 - iy1);
        for (int j = i + 1 + tid; j < NDET; j += NMS_T) {
            if (supp[j]) continue;
            float lx = fmaxf(ix1, sx1[j]);
            float ly = fmaxf(iy1, sy1[j]);
            float rx = fminf(ix2, sx2[j]);
            float ry = fminf(iy2, sy2[j]);
            float iw = fmaxf(rx - lx, 0.0f), ih = fmaxf(ry - ly, 0.0f);
            float inter = iw * ih;
            float ja = (sx2[j] - sx1[j]) * (sy2[j] - sy1[j]);
            float iou = inter / (iarea + ja - inter + 1e-7f);
            if (iou > IOU_T) supp[j] = 1;
        }
        __syncthreads();
    }
    __syncthreads();

    // stable compaction of kept boxes -> first KTOP slots
    if (tid == 0) {
        int cnt = 0;
        for (int i = 0; i < NDET && cnt < KTOP; ++i) {
            if (skey[i] > CONF_T && !supp[i]) {
                float* sb = selBoxes + ((size_t)batch * KTOP + cnt) * 4;
                sb[0] = sx1[i]; sb[1] = sy1[i]; sb[2] = sx2[i]; sb[3] = sy2[i];
                validW[batch * KTOP + cnt] = 1.0f;
                ++cnt;
            }
        }
        for (; cnt < KTOP; ++cnt) {
            float* sb = selBoxes + ((size_t)batch * KTOP + cnt) * 4;
            sb[0] = sb[1] = sb[2] = sb[3] = 0.0f;
            validW[batch * KTOP + cnt] = 0.0f;   // output zeroed for these slots
        }
    }
}

// ---------------------------------------------------------------------------
// Kernel 2: ROI pool gather -> padded crops [1600][160]
// one block (64 threads) per box
// ---------------------------------------------------------------------------
__global__ __launch_bounds__(64) void roi_kernel(
    const float* __restrict__ x,         // [16][3][640][640]
    const float* __restrict__ selBoxes,  // [1600][4]
    float* __restrict__ crops)           // [1600][160]
{
    const int bid   = blockIdx.x;        // 0..1599
    const int batch = bid / KTOP;
    const float* bb = selBoxes + (size_t)bid * 4;
    float x1 = bb[0], y1 = bb[1], x2 = bb[2], y2 = bb[3];

    for (int e = threadIdx.x; e < K1P; e += 64) {
        float val = 0.0f;
        if (e < 147) {
            int c = e / 49, rem = e % 49, py = rem / 7, px = rem % 7;
            float xs = x1 + (x2 - x1) * ((float)px + 0.5f) / 7.0f;
            float ys = y1 + (y2 - y1) * ((float)py + 0.5f) / 7.0f;
            int xi = (int)fminf(fmaxf(xs, 0.0f), 639.0f);
            int yi = (int)fminf(fmaxf(ys, 0.0f), 639.0f);
            val = x[(((size_t)batch * 3 + c) * 640 + yi) * 640 + xi];
        }
        crops[(size_t)bid * K1P + e] = val;
    }
}

// ---------------------------------------------------------------------------
// Kernel 3: fused 3-layer MLP via V_WMMA_F32_16X16X4_F32
// one block per 16-row stripe (grid=100); 8 waves (256 threads)
// ---------------------------------------------------------------------------
__global__ __launch_bounds__(256) void mlp_kernel(
    const float* __restrict__ crops,     // [1600][160]
    const float* __restrict__ W1p,       // [160][256]
    const float* __restrict__ W2p,       // [256][128]
    const float* __restrict__ W3p,       // [128][16]
    const float* __restrict__ b1p, const float* __restrict__ b2p,
    const float* __restrict__ b3p,
    const float* __restrict__ validW,    // [1600]
    float* __restrict__ out)             // [1600][4]
{
    __shared__ float As[16 * LDA];
    __shared__ float Fs[16 * LDF];
    __shared__ float Hs[16 * LDH];

    const int tid  = threadIdx.x;
    const int lane = tid & 31;
    const int wave = tid >> 5;
    const int r0   = blockIdx.x * 16;

    const int m      = lane & 15;              // row (A) / col (B,C/D) within tile
    const int kk     = (lane < 16) ? 0 : 2;    // K sub-offset per lane half
    const int rowOff = (lane < 16) ? 0 : 8;    // C/D row offset per lane half

    // stage 16 x 160 crop stripe
    for (int idx = tid; idx < 16 * K1P; idx += 256) {
        int r = idx / K1P, cc = idx % K1P;
        As[r * LDA + cc] = crops[((size_t)(r0 + r)) * K1P + cc];
    }
    __syncthreads();

    // ---- layer 1: 16x160 @ 160x256 ; each wave does cols [16w) and [128+16w) ----
    {
        const int n0a = wave * 16, n0b = 128 + wave * 16;
        v8f acc0 = {0.f,0.f,0.f,0.f,0.f,0.f,0.f,0.f};
        v8f acc1 = {0.f,0.f,0.f,0.f,0.f,0.f,0.f,0.f};
        for (int k0 = 0; k0 < K1P; k0 += 4) {
            v2f a;  a.x  = As[m * LDA + k0 + kk];        a.y  = As[m * LDA + k0 + kk + 1];
            v2f bA; bA.x = W1p[(k0 + kk) * N1P + n0a + m]; bA.y = W1p[(k0 + kk + 1) * N1P + n0a + m];
            v2f bB; bB.x = W1p[(k0 + kk) * N1P + n0b + m]; bB.y = W1p[(k0 + kk + 1) * N1P + n0b + m];
            acc0 = __builtin_amdgcn_wmma_f32_16x16x4_f32(false, a, false, bA, (short)0, acc0, false, false);
            acc1 = __builtin_amdgcn_wmma_f32_16x16x4_f32(false, a, false, bB, (short)0, acc1, false, false);
        }
        float biasA = b1p[n0a + m], biasB = b1p[n0b + m];
        #pragma unroll
        for (int i = 0; i < 8; ++i) {
            Fs[(i + rowOff) * LDF + n0a + m] = acc0[i] + biasA;
            Fs[(i + rowOff) * LDF + n0b + m] = acc1[i] + biasB;
        }
    }
    __syncthreads();

    // ---- layer 2: 16x256 @ 256x128 + ReLU ; wave -> cols [16w) ----
    {
        const int n0 = wave * 16;
        v8f acc = {0.f,0.f,0.f,0.f,0.f,0.f,0.f,0.f};
        for (int k0 = 0; k0 < N1P; k0 += 4) {
            v2f a; a.x = Fs[m * LDF + k0 + kk];          a.y = Fs[m * LDF + k0 + kk + 1];
            v2f b; b.x = W2p[(k0 + kk) * N2 + n0 + m];   b.y = W2p[(k0 + kk + 1) * N2 + n0 + m];
            acc = __builtin_amdgcn_wmma_f32_16x16x4_f32(false, a, false, b, (short)0, acc, false, false);
        }
        float bias = b2p[n0 + m];
        #pragma unroll
        for (int i = 0; i < 8; ++i)
            Hs[(i + rowOff) * LDH + n0 + m] = fmaxf(acc[i] + bias, 0.0f);
    }
    __syncthreads();

    // ---- layer 3: 16x128 @ 128x16 (only 4 cols valid) ; wave 0 only ----
    if (wave == 0) {
        v8f acc = {0.f,0.f,0.f,0.f,0.f,0.f,0.f,0.f};
        for (int k0 = 0; k0 < N2; k0 += 4) {
            v2f a; a.x = Hs[m * LDH + k0 + kk];          a.y = Hs[m * LDH + k0 + kk + 1];
            v2f b; b.x = W3p[(k0 + kk) * N3P + m];       b.y = W3p[(k0 + kk + 1) * N3P + m];
            acc = __builtin_amdgcn_wmma_f32_16x16x4_f32(false, a, false, b, (short)0, acc, false, false);
        }
        float bias = b3p[m];
        #pragma unroll
        for (int i = 0; i < 8; ++i) {
            int gr = r0 + i + rowOff;
            float v = acc[i] + bias;
            if (m < 4)
                out[(size_t)gr * 4 + m] = (validW[gr] > 0.5f) ? v : 0.0f;
        }
    }
}

// ---------------------------------------------------------------------------
extern "C" void kernel_launch(void* const* d_in, const int* in_sizes, int n_in,
                              void* d_out, int out_size, void* d_ws, size_t ws_size,
                              hipStream_t stream) {
    const float* x    = (const float*)d_in[0];   // [16][3][640][640]
    const float* dets = (const float*)d_in[1];   // [16][1024][6]
    const float* W1   = (const float*)d_in[2];   // [147][255]
    const float* b1   = (const float*)d_in[3];   // [255]
    const float* W2   = (const float*)d_in[4];   // [255][128]
    const float* b2   = (const float*)d_in[5];   // [128]
    const float* W3   = (const float*)d_in[6];   // [128][4]
    const float* b3   = (const float*)d_in[7];   // [4]
    float* out = (float*)d_out;                  // [16][100][4]

    float* ws = (float*)d_ws;
    float* W1p      = ws;                        // 40960
    float* W2p      = W1p + K1P * N1P;           // 32768
    float* W3p      = W2p + N1P * N2;            // 2048
    float* b1p      = W3p + N2 * N3P;            // 256
    float* b2p      = b1p + N1P;                 // 128
    float* b3p      = b2p + N2;                  // 16
    float* selBoxes = b3p + N3P;                 // 6400
    float* validW   = selBoxes + NBATCH * KTOP * 4; // 1600
    float* crops    = validW + NBATCH * KTOP;    // 256000

    const int packElems = K1P * N1P + N1P * N2 + N2 * N3P + N1P + N2 + N3P; // 76176
    pack_kernel<<<(packElems + 255) / 256, 256, 0, stream>>>(
        W1, b1, W2, b2, W3, b3, W1p, W2p, W3p, b1p, b2p, b3p);

    nms_kernel<<<NBATCH, NMS_T, 0, stream>>>(dets, selBoxes, validW);

    roi_kernel<<<NBATCH * KTOP, 64, 0, stream>>>(x, selBoxes, crops);

    mlp_kernel<<<(NBATCH * KTOP) / 16, 256, 0, stream>>>(
        crops, W1p, W2p, W3p, b1p, b2p, b3p, validW, out);
}